// VQCNet_23398981829250
// MI455X (gfx1250) — compile-verified
//
#include <hip/hip_runtime.h>

// ---------------------------------------------------------------------------
// Fused VQC net for MI455X (gfx1250, wave32):
//   pre = x @ W_pre.T + b_pre   (f32 WMMA 16x16x4; async global->LDS staging
//                                via GLOBAL_LOAD_ASYNC_TO_LDS_B128, double-
//                                buffered, ASYNCcnt-pipelined)
//   angles = tanh(pre) * pi     (hw exp based tanh)
//   4-qubit statevector sim     (register resident, packed-f32 complex)
//   out = q_out @ W_post.T + b_post
// Single pass over x (128 MB) -> ~5.6 us HBM floor @ 23.3 TB/s.
// ---------------------------------------------------------------------------

typedef __attribute__((ext_vector_type(2))) float v2f;
typedef __attribute__((ext_vector_type(4))) float v4f;
typedef __attribute__((ext_vector_type(8))) float v8f;

#define N_LAYERS 6
#define KDIM 512
#define KCHUNK 64
#define XROWSTRIDE 68              // dwords: 64 data + 4 pad (conflict-free b64 reads)
#define XTILE (16 * XROWSTRIDE)    // dwords per staging buffer
#define WPSTRIDE 516               // dwords: 512 data + 4 pad
#define ROWS_PER_BLOCK 128
#define BLOCK_THREADS 128
#define PI_F 3.14159265358979323846f

// Apply a 2x2 complex gate on qubit WIRE (bit 3-WIRE of flattened index).
// Amplitudes are float2 (re,im); gate values are duplicated pairs
// (gr,gr),(gi,gi) so each line maps to packed 2-wide f32 FMAs.
template <int WIRE>
__device__ __forceinline__ void apply_rot(v2f (&amp)[16], const float* __restrict__ gbase) {
  const v4f* G = reinterpret_cast<const v4f*>(gbase);
  const v4f G0 = G[0], G1 = G[1], G2 = G[2], G3 = G[3];   // 4x ds_load_b128
  const v2f g00r = __builtin_shufflevector(G0, G0, 0, 1);
  const v2f g00i = __builtin_shufflevector(G0, G0, 2, 3);
  const v2f g01r = __builtin_shufflevector(G1, G1, 0, 1);
  const v2f g01i = __builtin_shufflevector(G1, G1, 2, 3);
  const v2f g10r = __builtin_shufflevector(G2, G2, 0, 1);
  const v2f g10i = __builtin_shufflevector(G2, G2, 2, 3);
  const v2f g11r = __builtin_shufflevector(G3, G3, 0, 1);
  const v2f g11i = __builtin_shufflevector(G3, G3, 2, 3);
  constexpr int st = 1 << (3 - WIRE);
#pragma unroll
  for (int i = 0; i < 16; ++i) {
    if (i & st) continue;
    const int j = i + st;
    const v2f s0 = amp[i], s1 = amp[j];
    v2f s0n; s0n.x = -s0.y; s0n.y = s0.x;   // i*s0 (imag path of complex mult)
    v2f s1n; s1n.x = -s1.y; s1n.y = s1.x;
    amp[i] = s0 * g00r + s0n * g00i + s1 * g01r + s1n * g01i;
    amp[j] = s0 * g10r + s0n * g10i + s1 * g11r + s1n * g11i;
  }
}

__global__ __launch_bounds__(BLOCK_THREADS) void vqc_fused(
    const float* __restrict__ x, const float* __restrict__ W_pre,
    const float* __restrict__ b_pre, const float* __restrict__ q_w,
    const float* __restrict__ W_post, const float* __restrict__ b_post,
    float* __restrict__ out) {
  __shared__ __align__(16) float lds_x[4 * 2 * XTILE];         // per-wave double buffer
  __shared__ __align__(16) float lds_wp[5 * WPSTRIDE];         // rows 0-3 = W_pre, row 4 = zeros
  __shared__ __align__(16) float lds_g[N_LAYERS][4][16];       // gates, duplicated-pair form
  __shared__ __align__(16) float lds_ang[ROWS_PER_BLOCK][4];   // angles handoff

  const int tid = threadIdx.x;

  // ---- one-time LDS init (no div/mod) -------------------------------------
#pragma unroll
  for (int nn = 0; nn < 4; ++nn)
    for (int kk = tid; kk < WPSTRIDE; kk += BLOCK_THREADS)
      lds_wp[nn * WPSTRIDE + kk] = (kk < KDIM) ? W_pre[nn * KDIM + kk] : 0.0f;
  for (int kk = tid; kk < WPSTRIDE; kk += BLOCK_THREADS)
    lds_wp[4 * WPSTRIDE + kk] = 0.0f;

  if (tid < N_LAYERS * 4) {
    const int l = tid >> 2, w = tid & 3;
    const float phi = q_w[(l * 4 + w) * 3 + 0];
    const float th  = q_w[(l * 4 + w) * 3 + 1];
    const float om  = q_w[(l * 4 + w) * 3 + 2];
    const float c = cosf(0.5f * th), s = sinf(0.5f * th);
    const float app = 0.5f * (phi + om), apm = 0.5f * (phi - om);
    const float epr = cosf(app), epi = -sinf(app);
    const float emr = cosf(apm), emi = -sinf(apm);
    float* g = &lds_g[l][w][0];
    // g00 = ep*c ; g01 = -conj(em)*s ; g10 = em*s ; g11 = conj(ep)*c
    g[0]  = epr * c;   g[1]  = epr * c;     // g00r pair
    g[2]  = epi * c;   g[3]  = epi * c;     // g00i pair
    g[4]  = -emr * s;  g[5]  = -emr * s;    // g01r pair
    g[6]  = emi * s;   g[7]  = emi * s;     // g01i pair
    g[8]  = emr * s;   g[9]  = emr * s;     // g10r pair
    g[10] = emi * s;   g[11] = emi * s;     // g10i pair
    g[12] = epr * c;   g[13] = epr * c;     // g11r pair
    g[14] = -epi * c;  g[15] = -epi * c;    // g11i pair
  }
  __syncthreads();

  // ---- GEMM: pre = x @ W_pre.T  (one 16-row tile per wave) ----------------
  const int wave = tid >> 5;
  const int lane = tid & 31;
  const int n = lane & 15;   // A: row within tile / B,C: column N
  const int hi = lane >> 4;  // lane half selector
  const long tile_row0 = (long)blockIdx.x * ROWS_PER_BLOCK + (long)wave * 16;

  float* buf0 = lds_x + wave * (2 * XTILE);
  float* buf1 = buf0 + XTILE;
  // B operand base: lanes n<4 -> weight row n; lanes n>=4 -> zero row 4
  // (broadcast read, conflict-free). All per-iteration offsets are immediates.
  const int brow = (n < 4) ? n : 4;
  const float* bq = lds_wp + brow * WPSTRIDE + 2 * hi;
  const float* aptrs[2] = {buf0 + n * XROWSTRIDE + 2 * hi,
                           buf1 + n * XROWSTRIDE + 2 * hi};

  // Per-lane constant addresses for the async copies: LDS byte addresses
  // (low 32 bits of the flat pointer) and 32-bit global byte offsets.
  unsigned int ldsoff[2][8], goff[8];
#pragma unroll
  for (int j = 0; j < 8; ++j) {
    const int rr = 2 * j + hi;
    ldsoff[0][j] = (unsigned int)(uintptr_t)(buf0 + rr * XROWSTRIDE + n * 4);
    ldsoff[1][j] = (unsigned int)(uintptr_t)(buf1 + rr * XROWSTRIDE + n * 4);
    goff[j] = (unsigned int)(((tile_row0 + rr) * (long)KDIM + n * 4) * 4);
  }

  // Issue 8 async 16B/lane copies (one 16-row x 64-float chunk) into buffer ib.
  // GVS addressing: mem = SGPR64(x + 4*k0) + VGPR32(goff).  ASYNCcnt += 8.
  auto issue = [&](int ib, int k0) {
    const unsigned long long sb =
        (unsigned long long)(uintptr_t)x + (unsigned long long)(4 * k0);
#pragma unroll
    for (int j = 0; j < 8; ++j) {
      asm volatile("global_load_async_to_lds_b128 %0, %1, %2"
                   :: "v"(ldsoff[ib][j]), "v"(goff[j]), "s"(sb)
                   : "memory");
    }
  };

  v8f acc = {0.f, 0.f, 0.f, 0.f, 0.f, 0.f, 0.f, 0.f};

  issue(0, 0);
#pragma unroll
  for (int k0 = 0; k0 < KDIM; k0 += KCHUNK) {
    const int ib = (k0 / KCHUNK) & 1;
    const bool more = (k0 + KCHUNK) < KDIM;
    if (more) {
      issue(ib ^ 1, k0 + KCHUNK);   // prefetch next chunk into other buffer
      // 16 outstanding; ASYNCcnt is in-order -> <=8 means current chunk landed
      asm volatile("s_wait_asynccnt 0x8" ::: "memory");
    } else {
      asm volatile("s_wait_asynccnt 0x0" ::: "memory");
    }
    // 16 K-chunks of 4 (16x4 f32 A layout: lanes 0-15 K={0,1}, lanes 16-31
    // K={2,3}); B mirrors it. All LDS offsets are immediates.
    const float* aptr = aptrs[ib];
#pragma unroll
    for (int c = 0; c < KCHUNK / 4; ++c) {
      const v2f a = *reinterpret_cast<const v2f*>(aptr + 4 * c);
      const v2f b = *reinterpret_cast<const v2f*>(bq + 4 * c);
      acc = __builtin_amdgcn_wmma_f32_16x16x4_f32(false, a, false, b, (short)0, acc,
                                                  false, false);
    }
    bq += KCHUNK;
  }

  // Epilogue: C layout -> lane(l), VGPR r holds M = r + 8*(l>=16), N = l&15.
  if (n < 4) {
    const float bp = b_pre[n];
#pragma unroll
    for (int r = 0; r < 8; ++r) {
      const int m = r + 8 * hi;
      const float p2 = acc[r] + bp;
      const float t = __expf(2.0f * p2);                  // tanh via hw exp
      lds_ang[wave * 16 + m][n] = __fdividef(t - 1.0f, t + 1.0f) * PI_F;
    }
  }
  __syncthreads();

  // ---- quantum sim: one sample per thread (all 128 threads) ---------------
  {
    v2f amp[16];
    // RY embedding from |0000>: real product state
    // amp[i] = prod_w (bit_w(i) ? sin : cos)(a_w/2),  bit_w = bit (3-w) of i.
    const v4f av = *reinterpret_cast<const v4f*>(&lds_ang[tid][0]);
    const float c0 = __cosf(0.5f * av.x), s0 = __sinf(0.5f * av.x);
    const float c1 = __cosf(0.5f * av.y), s1 = __sinf(0.5f * av.y);
    const float c2 = __cosf(0.5f * av.z), s2 = __sinf(0.5f * av.z);
    const float c3 = __cosf(0.5f * av.w), s3 = __sinf(0.5f * av.w);
    const float f01[4] = {c0 * c1, c0 * s1, s0 * c1, s0 * s1};
    const float f23[4] = {c2 * c3, c2 * s3, s2 * c3, s2 * s3};
#pragma unroll
    for (int i = 0; i < 16; ++i) {
      amp[i].x = f01[i >> 2] * f23[i & 3];
      amp[i].y = 0.f;
    }

#pragma unroll
    for (int l = 0; l < N_LAYERS; ++l) {
      apply_rot<0>(amp, &lds_g[l][0][0]);
      apply_rot<1>(amp, &lds_g[l][1][0]);
      apply_rot<2>(amp, &lds_g[l][2][0]);
      apply_rot<3>(amp, &lds_g[l][3][0]);
      const int rsh = l % 3 + 1;
#pragma unroll
      for (int w = 0; w < 4; ++w) {
        const int cb = 1 << (3 - w);
        const int tb = 1 << (3 - ((w + rsh) & 3));
#pragma unroll
        for (int i = 0; i < 16; ++i) {
          if ((i & cb) && !(i & tb)) {
            const int j = i | tb;
            const v2f t = amp[i]; amp[i] = amp[j]; amp[j] = t;
          }
        }
      }
    }

    // Pauli-Z expectations
    float e0 = 0.f, e1 = 0.f, e2 = 0.f, e3 = 0.f;
#pragma unroll
    for (int i = 0; i < 16; ++i) {
      const float p = amp[i].x * amp[i].x + amp[i].y * amp[i].y;
      e0 += (i & 8) ? -p : p;
      e1 += (i & 4) ? -p : p;
      e2 += (i & 2) ? -p : p;
      e3 += (i & 1) ? -p : p;
    }

    // out = q_out @ W_post.T + b_post (W_post/b_post uniform -> scalar loads)
    const long row = (long)blockIdx.x * ROWS_PER_BLOCK + tid;
#pragma unroll
    for (int cc = 0; cc < 10; ++cc) {
      const float o = b_post[cc] + e0 * W_post[cc * 4 + 0] + e1 * W_post[cc * 4 + 1] +
                      e2 * W_post[cc * 4 + 2] + e3 * W_post[cc * 4 + 3];
      __builtin_nontemporal_store(o, &out[row * 10 + cc]);
    }
  }
}

extern "C" void kernel_launch(void* const* d_in, const int* in_sizes, int n_in,
                              void* d_out, int out_size, void* d_ws, size_t ws_size,
                              hipStream_t stream) {
  const float* x = (const float*)d_in[0];
  const float* W_pre = (const float*)d_in[1];
  const float* b_pre = (const float*)d_in[2];
  const float* q_w = (const float*)d_in[3];
  const float* W_post = (const float*)d_in[4];
  const float* b_post = (const float*)d_in[5];
  float* out = (float*)d_out;

  const int nrows = in_sizes[0] / KDIM;           // 65536
  const int grid = nrows / ROWS_PER_BLOCK;        // 512
  vqc_fused<<<grid, BLOCK_THREADS, 0, stream>>>(x, W_pre, b_pre, q_w, W_post, b_post, out);
}